// KANActivation_13408887899047
// MI455X (gfx1250) — compile-verified
//
#include <hip/hip_runtime.h>

// KAN activation for MI455X (gfx1250, wave32).
// out[b,o,i] = sum_d bspline3(x[b,i])_d * coef[o,i,d]
// 128 independent (1024x11)*(11x128) GEMMs via V_WMMA_F32_16X16X4_F32 (exact f32).
//
// Block = (batch-tile of 16) x (group of 8 i). 8 waves/block; wave w owns
// i = igrp*8 + w, computes the 16 spline rows once (2x redundancy across
// half-waves only), then loops the 8 o-tiles reusing the A operands.

typedef __attribute__((ext_vector_type(2))) float v2f;
typedef __attribute__((ext_vector_type(8))) float v8f;

#define KAN_BATCH   1024
#define KAN_IN      128
#define KAN_OUT     128
#define KAN_NCOEF   11      // GRID_SIZE + SPLINE_ORDER
#define KAN_NKNOT   15      // GRID_SIZE + 2*SPLINE_ORDER + 1

__global__ __launch_bounds__(256) void kan_wmma_f32_kernel(
    const float* __restrict__ x,     // (BATCH, IN)
    const float* __restrict__ grid,  // (OUT, IN, NKNOT) broadcast; first 15 = knots
    const float* __restrict__ coef,  // (OUT, IN, NCOEF)
    float* __restrict__ out)         // (BATCH, OUT, IN)
{
    const int tid  = threadIdx.x;
    const int lane = tid & 31;
    const int wave = tid >> 5;          // 0..7
    const int lo   = lane & 15;
    const int hi   = lane >> 4;
    const bool hib = (hi != 0);

    const int btile = blockIdx.x >> 4;  // 0..63
    const int igrp  = blockIdx.x & 15;  // 0..15
    const int i     = igrp * 8 + wave;  // this wave's input channel

    // ---- knots (lane-uniform -> scalar loads) ----
    float gk[KAN_NKNOT];
#pragma unroll
    for (int j = 0; j < KAN_NKNOT; ++j) gk[j] = grid[j];

    // ---- hoisted reciprocals of knot spans (lane-uniform, v_rcp_f32) ----
    float rd1[14], rd2[13], rd3[12];
#pragma unroll
    for (int j = 0; j < 14; ++j) rd1[j] = __builtin_amdgcn_rcpf(gk[j + 1] - gk[j]);
#pragma unroll
    for (int j = 0; j < 13; ++j) rd2[j] = __builtin_amdgcn_rcpf(gk[j + 2] - gk[j]);
#pragma unroll
    for (int j = 0; j < 12; ++j) rd3[j] = __builtin_amdgcn_rcpf(gk[j + 3] - gk[j]);

    // ---- per-lane x value: row b = 16*btile + (lane&15) ----
    const int brow = btile * 16 + lo;
    const float xv = x[brow * KAN_IN + i];

    // ---- Cox-de Boor, order 3, fully unrolled, constant indices only ----
    float bc[14];
#pragma unroll
    for (int j = 0; j < 14; ++j)
        bc[j] = (xv >= gk[j] && xv < gk[j + 1]) ? 1.0f : 0.0f;
#pragma unroll
    for (int j = 0; j < 13; ++j)      // degree 1 (ascending j reads old j+1)
        bc[j] = (xv - gk[j]) * rd1[j] * bc[j] + (gk[j + 2] - xv) * rd1[j + 1] * bc[j + 1];
#pragma unroll
    for (int j = 0; j < 12; ++j)      // degree 2
        bc[j] = (xv - gk[j]) * rd2[j] * bc[j] + (gk[j + 3] - xv) * rd2[j + 1] * bc[j + 1];
#pragma unroll
    for (int j = 0; j < 11; ++j)      // degree 3 -> bc[0..10] are the 11 bases
        bc[j] = (xv - gk[j]) * rd3[j] * bc[j] + (gk[j + 4] - xv) * rd3[j + 1] * bc[j + 1];

    // ---- A operands in WMMA layout (16x4 f32: M=lane&15; lo half K={4c,4c+1},
    //      hi half K={4c+2,4c+3}); constant indices + one select on hi ----
    v2f Av[3];
#pragma unroll
    for (int c = 0; c < 3; ++c) {
        const float e0 = bc[4 * c + 0];
        const float e1 = bc[4 * c + 1];
        const float e2 = bc[4 * c + 2];
        const float e3 = (4 * c + 3 < KAN_NCOEF) ? bc[4 * c + 3] : 0.0f;  // pad K=11
        Av[c].x = hib ? e2 : e0;
        Av[c].y = hib ? e3 : e1;
    }

    // ---- loop the 8 o-tiles, reusing Av ----
    for (int ot = 0; ot < 8; ++ot) {
        const int ocol = ot * 16 + lo;                       // N = lane&15
        const float* cpb = coef + (ocol * KAN_IN + i) * KAN_NCOEF;
        const float* cp  = cpb + 2 * hi;                     // K-half offset

        v2f Bm0, Bm1, Bm2;
        Bm0.x = cp[0];  Bm0.y = cp[1];
        Bm1.x = cp[4];  Bm1.y = cp[5];
        Bm2.x = cp[8];
        const float t9 = cpb[9];                             // always in-bounds
        Bm2.y = hib ? 0.0f : t9;                             // K=11 pad for hi half

        v8f acc = {};
        acc = __builtin_amdgcn_wmma_f32_16x16x4_f32(false, Av[0], false, Bm0,
                                                    (short)0, acc, false, false);
        acc = __builtin_amdgcn_wmma_f32_16x16x4_f32(false, Av[1], false, Bm1,
                                                    (short)0, acc, false, false);
        acc = __builtin_amdgcn_wmma_f32_16x16x4_f32(false, Av[2], false, Bm2,
                                                    (short)0, acc, false, false);

        // D tile: lane holds col N = lane&15, rows M = r + 8*hi
#pragma unroll
        for (int r = 0; r < 8; ++r) {
            const int b = btile * 16 + r + 8 * hi;
            out[(b * KAN_OUT + ocol) * KAN_IN + i] = acc[r];
        }
    }
}

extern "C" void kernel_launch(void* const* d_in, const int* in_sizes, int n_in,
                              void* d_out, int out_size, void* d_ws, size_t ws_size,
                              hipStream_t stream) {
    const float* x    = (const float*)d_in[0];  // (1024, 128)
    const float* grid = (const float*)d_in[1];  // (128, 128, 15) broadcast knots
    const float* coef = (const float*)d_in[2];  // (128, 128, 11)
    float* out = (float*)d_out;                 // (1024, 128, 128)

    const int nblocks = (KAN_BATCH / 16) * (KAN_IN / 8);  // 64 * 16 = 1024
    kan_wmma_f32_kernel<<<nblocks, 256, 0, stream>>>(x, grid, coef, out);
}